// Net_44976897524569
// MI455X (gfx1250) — compile-verified
//
#include <hip/hip_runtime.h>
#include <hip/hip_bf16.h>

typedef __attribute__((ext_vector_type(2))) float v2f;
typedef __attribute__((ext_vector_type(8))) float v8f;

// ---------------------------------------------------------------------------
// K1: deg[v] = 1.0 (self-loop).
__global__ void k_init_deg(float* __restrict__ deg, int n) {
    int i = blockIdx.x * blockDim.x + threadIdx.x;
    if (i < n) deg[i] = 1.0f;
}

// K2: in-degree count over real edges (float atomics; exact for counts < 2^24)
__global__ void k_count_deg(const int* __restrict__ dst, float* __restrict__ deg, int E) {
    int e = blockIdx.x * blockDim.x + threadIdx.x;
    if (e < E) atomicAdd(&deg[dst[e]], 1.0f);
}

// K3: dinv = rsqrt(deg) (in place), p = x*dinv, s1acc seeded with self-loop p[v].
__global__ void k_node1(const float* __restrict__ x, float* __restrict__ deg_dinv,
                        float* __restrict__ p, float* __restrict__ s1acc, int n) {
    int i = blockIdx.x * blockDim.x + threadIdx.x;
    if (i < n) {
        float dv = rsqrtf(deg_dinv[i]);   // deg >= 1 always (self-loop)
        deg_dinv[i] = dv;
        float pv = x[i] * dv;
        p[i] = pv;
        s1acc[i] = pv;
    }
}

// K4: layer-1 aggregate: s1acc[dst] += p[src]
__global__ void k_scatter1(const int* __restrict__ src, const int* __restrict__ dst,
                           const float* __restrict__ p, float* __restrict__ s1acc, int E) {
    int e = blockIdx.x * blockDim.x + threadIdx.x;
    if (e < E) atomicAdd(&s1acc[dst[e]], p[src[e]]);
}

// ---------------------------------------------------------------------------
// K5 (WMMA): per 16-node tile:
//   s1[v] = dinv[v]*s1acc[v]
//   z[v]  = sum_j W2[j]*relu(W1[j]*s1[v] + b1[j])   -> 16x accumulating
//           V_WMMA_F32_16X16X4_F32, A[m,k]=relu(W1[4c+k]*s1[m]+b1[4c+k]),
//           B column 0 = W2 chunk (other columns zero).
//   q[v]  = z[v]*dinv[v]; store q, seed outacc[v]=q[v] (layer-2 self-loop).
//
// Weights staged in 768B of LDS once per block (ds_load_2addr_b64 broadcast in
// the loop, no bank conflicts, no SGPR pressure). EXEC is all-ones at every
// WMMA (branchless operand setup). Result extraction goes through a per-wave
// 16-float LDS staging buffer: lanes 0/16 dump their 8 accumulator VGPRs with
// two ds_store_b128 each, then lanes 0-15 read back and issue one coalesced
// dinv load + two coalesced global stores under a single guard (instead of 16
// divergent exec-save regions).
__global__ void __launch_bounds__(256) k_z_wmma(
        const float* __restrict__ s1acc, const float* __restrict__ dinv,
        const float* __restrict__ W1, const float* __restrict__ b1,
        const float* __restrict__ W2,
        float* __restrict__ q, float* __restrict__ outacc, int n) {
    __shared__ __align__(16) float sW1[64];
    __shared__ __align__(16) float sB1[64];
    __shared__ __align__(16) float sW2[64];
    __shared__ __align__(16) float sZ[8][16];      // per-wave z staging

    const int t = threadIdx.x;
    if (t < 64) {                     // cooperative one-shot weight stage
        sW1[t] = W1[t];
        sB1[t] = b1[t];
        sW2[t] = W2[t];
    }
    __syncthreads();

    const int lane = t & 31;
    const int wave = t >> 5;
    const int tile = blockIdx.x * (blockDim.x >> 5) + wave;
    const int base = tile * 16;
    const int m    = lane & 15;

    int node = base + m;
    int cl   = node < n ? node : (n - 1);          // branchless clamp, EXEC stays full
    float s1 = dinv[cl] * s1acc[cl];
    s1 = (node < n) ? s1 : 0.0f;

    const int  k0   = (lane >> 4) << 1;            // 0 for lanes 0-15, 2 for 16-31
    const bool isN0 = (m == 0);                    // B column 0 only

    v8f acc = {};
    #pragma unroll 4
    for (int c = 0; c < 16; ++c) {
        const int kk = 4 * c + k0;                 // 8B-aligned pair offset
        float2 w1p = *(const float2*)&sW1[kk];     // ds broadcast loads
        float2 b1p = *(const float2*)&sB1[kk];
        float2 w2p = *(const float2*)&sW2[kk];

        v2f a, b;
        a.x = fmaxf(fmaf(w1p.x, s1, b1p.x), 0.0f); // relu(W1*s1 + b1)
        a.y = fmaxf(fmaf(w1p.y, s1, b1p.y), 0.0f);
        b.x = isN0 ? w2p.x : 0.0f;                 // W2 in column N=0 only
        b.y = isN0 ? w2p.y : 0.0f;
        // 8 args: (neg_a, A, neg_b, B, c_mod, C, reuse_a, reuse_b)
        acc = __builtin_amdgcn_wmma_f32_16x16x4_f32(
                  false, a, false, b, (short)0, acc, false, false);
    }

    // D column 0 lives in lane 0 (M=0..7 in acc[0..7]) and lane 16 (M=8..15).
    // Stage through LDS: one short writer branch, then coalesced stores.
    if (m == 0) {
        const int off = (lane >> 1) & 8;           // lane0 -> 0, lane16 -> 8
        float4 lo = make_float4(acc[0], acc[1], acc[2], acc[3]);
        float4 hi = make_float4(acc[4], acc[5], acc[6], acc[7]);
        *(float4*)&sZ[wave][off]     = lo;         // ds_store_b128 x2
        *(float4*)&sZ[wave][off + 4] = hi;
    }
    __syncthreads();

    if (lane < 16 && node < n) {
        float qv = sZ[wave][m] * dinv[node];       // coalesced load + stores
        q[node]      = qv;
        outacc[node] = qv;
    }
}

// K6: layer-2 aggregate: outacc[dst] += q[src]
__global__ void k_scatter2(const int* __restrict__ src, const int* __restrict__ dst,
                           const float* __restrict__ qv, float* __restrict__ outacc, int E) {
    int e = blockIdx.x * blockDim.x + threadIdx.x;
    if (e < E) atomicAdd(&outacc[dst[e]], qv[src[e]]);
}

// K7: out[v] = dinv[v]*outacc[v] + b2
__global__ void k_finish(const float* __restrict__ dinv, const float* __restrict__ outacc,
                         const float* __restrict__ b2, float* __restrict__ out, int n) {
    int i = blockIdx.x * blockDim.x + threadIdx.x;
    if (i < n) out[i] = dinv[i] * outacc[i] + b2[0];
}

// ---------------------------------------------------------------------------
extern "C" void kernel_launch(void* const* d_in, const int* in_sizes, int n_in,
                              void* d_out, int out_size, void* d_ws, size_t ws_size,
                              hipStream_t stream) {
    const float* x  = (const float*)d_in[0];   // [N,1]
    const int*   ei = (const int*)  d_in[1];   // [2,E] row-major: src then dst
    const float* W1 = (const float*)d_in[2];   // [1,64]
    const float* b1 = (const float*)d_in[3];   // [64]
    const float* W2 = (const float*)d_in[4];   // [64,1]
    const float* b2 = (const float*)d_in[5];   // [1]

    const int n = in_sizes[0];
    const int E = in_sizes[1] / 2;
    const int* src = ei;
    const int* dst = ei + E;

    float* ws      = (float*)d_ws;             // 5*n floats of scratch
    float* deg     = ws;                       // becomes dinv in place
    float* p       = ws + (size_t)n;
    float* s1acc   = ws + 2 * (size_t)n;
    float* q       = ws + 3 * (size_t)n;
    float* outacc  = ws + 4 * (size_t)n;
    float* out     = (float*)d_out;

    const int TB = 256;
    const int gn = (n + TB - 1) / TB;
    const int ge = (E + TB - 1) / TB;

    k_init_deg<<<gn, TB, 0, stream>>>(deg, n);
    k_count_deg<<<ge, TB, 0, stream>>>(dst, deg, E);
    k_node1<<<gn, TB, 0, stream>>>(x, deg, p, s1acc, n);
    k_scatter1<<<ge, TB, 0, stream>>>(src, dst, p, s1acc, E);

    const int tiles = (n + 15) / 16;           // 16 nodes per wave
    const int wavesPerBlock = TB / 32;         // 8 waves/block (wave32)
    const int gz = (tiles + wavesPerBlock - 1) / wavesPerBlock;
    k_z_wmma<<<gz, TB, 0, stream>>>(s1acc, deg, W1, b1, W2, q, outacc, n);

    k_scatter2<<<ge, TB, 0, stream>>>(src, dst, q, outacc, E);
    k_finish<<<gn, TB, 0, stream>>>(deg, outacc, b2, out, n);
}